// Prior_28312424415544
// MI455X (gfx1250) — compile-verified
//
#include <hip/hip_runtime.h>
#include <hip/hip_bf16.h>

typedef __attribute__((ext_vector_type(2))) float v2f;
typedef __attribute__((ext_vector_type(4))) float v4f;
typedef __attribute__((ext_vector_type(8))) float v8f;

// ---------------------------------------------------------------------------
// Problem constants (from the reference)
// ---------------------------------------------------------------------------
#define ZDIM   64
#define RANK   8
#define BATCH  8192
#define NCASE  6          // 2 causal (e) + 4 spurious (y*2+e)
#define LOG2PI 1.8378770664093454f

struct MlpW { const float *W1, *b1, *W2, *b2, *W3, *b3; };

// ---------------------------------------------------------------------------
// Setup: per case, run the 3 MLPs (one-hot input => first layer is a row
// select), build cov = LR*LR^T + diag(softplus(d)), Cholesky, logdet,
// Linv = L^-1 (lower-tri), u = Linv * mu.  6 blocks, 256 threads each.
// ---------------------------------------------------------------------------
__device__ inline void mlp_fwd(const MlpW& m, int row0, int row1, int dout,
                               float* h1, float* h2, float* outp) {
    const int tid = threadIdx.x;
    float v = m.b1[tid] + m.W1[row0 * 256 + tid];
    if (row1 >= 0) v += m.W1[row1 * 256 + tid];
    h1[tid] = fmaxf(v, 0.0f);
    __syncthreads();
    float a2 = m.b2[tid];
    for (int k = 0; k < 256; ++k) a2 += h1[k] * m.W2[k * 256 + tid];
    h2[tid] = fmaxf(a2, 0.0f);
    __syncthreads();
    for (int o = tid; o < dout; o += 256) {
        float a3 = m.b3[o];
        for (int k = 0; k < 256; ++k) a3 += h2[k] * m.W3[k * dout + o];
        outp[o] = a3;
    }
    __syncthreads();
}

__global__ __launch_bounds__(256)
void prior_setup(MlpW muc, MlpW lrc, MlpW dic,
                 MlpW mus, MlpW lrs, MlpW dis,
                 float* __restrict__ LinvAll,   // [6][64][64]
                 float* __restrict__ uAll,      // [6][64]
                 float* __restrict__ ldAll) {   // [6]
    __shared__ float h1[256], h2[256];
    __shared__ float muv[ZDIM], dgv[ZDIM], lrv[ZDIM * RANK];
    __shared__ float Am[ZDIM * ZDIM];   // cov -> Cholesky L (in place, lower)
    __shared__ float Lv[ZDIM * ZDIM];   // Linv
    __shared__ float sD;

    const int tid = threadIdx.x;
    const int cse = blockIdx.x;         // 0..1 causal (e), 2..5 spurious (y*2+e)

    int row0, row1; MlpW m0, m1, m2;
    if (cse < 2) { row0 = cse; row1 = -1; m0 = muc; m1 = lrc; m2 = dic; }
    else {
        int idx = cse - 2; int yy = idx >> 1; int ee = idx & 1;
        row0 = yy; row1 = 2 + ee;               // concat(y1h, e1h) => two rows
        m0 = mus; m1 = lrs; m2 = dis;
    }

    mlp_fwd(m0, row0, row1, ZDIM,        h1, h2, muv);
    mlp_fwd(m1, row0, row1, ZDIM * RANK, h1, h2, lrv);
    mlp_fwd(m2, row0, row1, ZDIM,        h1, h2, dgv);

    // cov = lr @ lr^T + diag(softplus(dg))
    for (int idx = tid; idx < ZDIM * ZDIM; idx += 256) {
        int i = idx >> 6, j = idx & 63;
        float s = 0.0f;
        #pragma unroll
        for (int r = 0; r < RANK; ++r) s += lrv[i * RANK + r] * lrv[j * RANK + r];
        if (i == j) {
            float x = dgv[i];
            float sp = (x > 0.0f) ? (x + log1pf(expf(-x))) : log1pf(expf(x));
            s += sp;
        }
        Am[idx] = s;
    }
    __syncthreads();

    // Cholesky (right-looking), lower triangle of Am
    for (int j = 0; j < ZDIM; ++j) {
        if (tid == 0) sD = sqrtf(Am[j * 64 + j]);
        __syncthreads();
        float Ljj = sD;
        if (tid < ZDIM) {
            if (tid == j)      Am[j * 64 + j] = Ljj;
            else if (tid > j)  Am[tid * 64 + j] /= Ljj;
        }
        __syncthreads();
        for (int idx = tid; idx < ZDIM * ZDIM; idx += 256) {
            int i = idx >> 6, k = idx & 63;
            if (i > j && k > j && k <= i)
                Am[idx] -= Am[i * 64 + j] * Am[k * 64 + j];
        }
        __syncthreads();
    }

    if (tid == 0) {
        float ld = 0.0f;
        for (int i = 0; i < ZDIM; ++i) ld += logf(Am[i * 64 + i]);
        ldAll[cse] = 2.0f * ld;
    }

    // Linv: forward-substitute L * X = I, thread c owns column c
    if (tid < ZDIM) {
        int c = tid;
        for (int i = 0; i < ZDIM; ++i) {
            if (i < c) { Lv[i * 64 + c] = 0.0f; continue; }
            float s = (i == c) ? 1.0f : 0.0f;
            for (int k = c; k < i; ++k) s -= Am[i * 64 + k] * Lv[k * 64 + c];
            Lv[i * 64 + c] = s / Am[i * 64 + i];
        }
    }
    __syncthreads();

    // u = Linv @ mu (row tid), and spill results
    if (tid < ZDIM) {
        float s = 0.0f;
        for (int k = 0; k <= tid; ++k) s += Lv[tid * 64 + k] * muv[k];
        uAll[cse * ZDIM + tid] = s;
    }
    for (int idx = tid; idx < ZDIM * ZDIM; idx += 256)
        LinvAll[cse * ZDIM * ZDIM + idx] = Lv[idx];
}

// ---------------------------------------------------------------------------
// Main: each wave handles 16 samples.  GEMM mapping: M = Linv rows, N =
// samples, K = 64 (16 chained V_WMMA_F32_16X16X4_F32).  With samples on the
// N axis, each lane owns ONE sample and eight Linv rows per M-tile, so the
// sum_i (t[i]-u[i])^2 reduction is pure in-lane VALU (no DS-port shuffles);
// a single xor-16 shuffle at the very end combines the two row-halves.
// A fragments (Linv) are preloaded into a register array so the 16 B64 loads
// pipeline instead of a wait per WMMA.
// ---------------------------------------------------------------------------
__global__ __launch_bounds__(128)
void prior_main(const int* __restrict__ yIn, const int* __restrict__ eIn,
                const float* __restrict__ z,
                const float* __restrict__ LinvAll,
                const float* __restrict__ uAll,
                const float* __restrict__ ldAll,
                float* __restrict__ out) {
    const int tid  = threadIdx.x;
    const int wave = tid >> 5;
    const int l    = tid & 31;
    const int hf   = l >> 4;       // K-half for A/B frags; row-half for D
    const int nlan = l & 15;       // N index: sample-in-tile (and M idx for A)

    const int w = blockIdx.x * 4 + wave;     // 512 waves total
    const int s = w * 16;                    // base sample of this wave's tile

    // this lane's sample (same for both row-halves)
    const int smp = s + nlan;
    const int eLane = eIn[smp];
    const int yLane = yIn[smp];
    const int tgtC  = eLane;                 // causal case id
    const int tgtS  = 2 + 2 * yLane + eLane; // spurious case id

    float maha = 0.0f;

    #pragma unroll
    for (int ph = 0; ph < 2; ++ph) {
        const int zoff = ph * 64;
        const int c0 = ph ? 2 : 0;
        const int c1 = ph ? 6 : 2;
        const int tgt = ph ? tgtS : tgtC;

        // B fragments (z): lane holds z[s+nlan][zoff + 4*kk + 2*hf .. +1]
        v2f bz[16];
        const float* zrow = z + (size_t)(s + nlan) * 128 + zoff + 2 * hf;
        #pragma unroll
        for (int kk = 0; kk < 16; ++kk)
            bz[kk] = *reinterpret_cast<const v2f*>(zrow + kk * 4);

        for (int c = c0; c < c1; ++c) {
            const float* Lc = LinvAll + c * (ZDIM * ZDIM);
            const float* uc = uAll + c * ZDIM;
            #pragma unroll
            for (int mt = 0; mt < 4; ++mt) {
                // A fragments (Linv rows mt*16..mt*16+15): preload all 16
                v2f af[16];
                const float* Arow = Lc + (mt * 16 + nlan) * 64 + 2 * hf;
                #pragma unroll
                for (int kk = 0; kk < 16; ++kk)
                    af[kk] = *reinterpret_cast<const v2f*>(Arow + kk * 4);

                // u for the 8 rows this lane reduces: contiguous, per-lane
                const float* up = uc + mt * 16 + 8 * hf;
                v4f u0 = *reinterpret_cast<const v4f*>(up);
                v4f u1 = *reinterpret_cast<const v4f*>(up + 4);

                v8f acc = {};
                #pragma unroll
                for (int kk = 0; kk < 16; ++kk)
                    acc = __builtin_amdgcn_wmma_f32_16x16x4_f32(
                        false, af[kk], false, bz[kk], (short)0, acc,
                        false, false);

                // in-lane: t = sum_j (acc[j] - u[j])^2  (rows mt*16+8*hf+j)
                float t = 0.0f;
                #pragma unroll
                for (int j = 0; j < 4; ++j) {
                    float d = acc[j] - u0[j];
                    t = fmaf(d, d, t);
                }
                #pragma unroll
                for (int j = 0; j < 4; ++j) {
                    float d = acc[4 + j] - u1[j];
                    t = fmaf(d, d, t);
                }
                maha += (tgt == c) ? t : 0.0f;
            }
        }
    }

    // combine the two row-halves of this lane's sample (one shuffle total)
    maha += __shfl_xor(maha, 16, 32);

    if (hf == 0) {
        float ld = ldAll[tgtC] + ldAll[tgtS];
        out[smp] = -0.5f * (128.0f * LOG2PI + ld + maha);
    }
}

// ---------------------------------------------------------------------------
// Launch.  Input order assumed = setup_inputs() insertion order:
//   0:y 1:e 2:z, then params {mu_c, lr_c, diag_c, mu_s, lr_s, diag_s},
//   each MLP flattened as W1,b1,W2,b2,W3,b3.
// Workspace: Linv[6][64][64] + u[6][64] + logdet[6]  (~100 KB, f32)
// ---------------------------------------------------------------------------
extern "C" void kernel_launch(void* const* d_in, const int* in_sizes, int n_in,
                              void* d_out, int out_size, void* d_ws, size_t ws_size,
                              hipStream_t stream) {
    (void)in_sizes; (void)n_in; (void)out_size; (void)ws_size;
    const int*   yIn = (const int*)d_in[0];
    const int*   eIn = (const int*)d_in[1];
    const float* z   = (const float*)d_in[2];

    auto F = [&](int i) { return (const float*)d_in[i]; };
    MlpW muc{F(3),  F(4),  F(5),  F(6),  F(7),  F(8)};
    MlpW lrc{F(9),  F(10), F(11), F(12), F(13), F(14)};
    MlpW dic{F(15), F(16), F(17), F(18), F(19), F(20)};
    MlpW mus{F(21), F(22), F(23), F(24), F(25), F(26)};
    MlpW lrs{F(27), F(28), F(29), F(30), F(31), F(32)};
    MlpW dis{F(33), F(34), F(35), F(36), F(37), F(38)};

    float* ws      = (float*)d_ws;
    float* LinvAll = ws;
    float* uAll    = ws + NCASE * ZDIM * ZDIM;
    float* ldAll   = uAll + NCASE * ZDIM;

    prior_setup<<<NCASE, 256, 0, stream>>>(muc, lrc, dic, mus, lrs, dis,
                                           LinvAll, uAll, ldAll);

    // 512 waves * 16 samples = 8192; 4 waves per block -> 128 blocks
    prior_main<<<BATCH / 64, 128, 0, stream>>>(yIn, eIn, z, LinvAll, uAll,
                                               ldAll, (float*)d_out);
}